// Encoder_51513837748917
// MI455X (gfx1250) — compile-verified
//
#include <hip/hip_runtime.h>
#include <hip/hip_bf16.h>

// GCN 2-layer forward for MI455X (gfx1250, wave32).
// GEMMs via V_WMMA_F32_16X16X4_F32 (full fp32); edge aggregation via
// f32 global atomics into an L2-resident (25.6MB << 192MB) accumulator.

typedef float v2f __attribute__((ext_vector_type(2)));
typedef float v8f __attribute__((ext_vector_type(8)));

#define IN_C  128
#define OUT1  64
#define OUT2  64

// ---------------- degree / normalization ----------------

__global__ __launch_bounds__(256) void k_fill1(float* __restrict__ deg, int n) {
    int i = blockIdx.x * blockDim.x + threadIdx.x;
    if (i < n) deg[i] = 1.0f;  // self-loop contributes 1 to every node's degree
}

__global__ __launch_bounds__(256) void k_deg_edges(const long long* __restrict__ ei,
                                                   float* __restrict__ deg, int E) {
    int e = blockIdx.x * blockDim.x + threadIdx.x;
    if (e < E) {
        long long d = ei[(size_t)E + e];   // dst row of edge_index
        atomicAdd(&deg[d], 1.0f);
    }
}

__global__ __launch_bounds__(256) void k_rsqrt(float* __restrict__ dinv, int n) {
    int i = blockIdx.x * blockDim.x + threadIdx.x;
    if (i < n) dinv[i] = rsqrtf(dinv[i]);  // deg >= 1 always (self-loop)
}

// ---------------- fp32 WMMA GEMM: H[N x 64] = X[N x K] @ W[K x 64] ----------------
// Block = 128 threads = 4 waves; block computes a 16-row x 64-col tile.
// Wave w computes columns [16w, 16w+16). K-loop in steps of 4 (16x16x4 f32 WMMA).
//
// A 16x4 f32 frag (2 VGPRs/lane): lanes 0-15 -> K={0,1}, lanes 16-31 -> K={2,3}.
// B 4x16 frag mirrored. C/D: VGPR r, lanes 0-15 -> M=r, lanes 16-31 -> M=8+r.

template <int K>
__global__ __launch_bounds__(128) void k_gemm_wmma(const float* __restrict__ X,
                                                   const float* __restrict__ W,
                                                   float* __restrict__ H, int nrows) {
    const int wave = threadIdx.x >> 5;
    const int lane = threadIdx.x & 31;
    const int half = lane >> 4;        // 0: lanes 0-15, 1: lanes 16-31
    const int l16  = lane & 15;
    const int koff = half * 2;

    int row = blockIdx.x * 16 + l16;
    if (row >= nrows) row = nrows - 1;          // clamp (EXEC must stay all-1 for WMMA)
    const int col = wave * 16 + l16;

    const float* __restrict__ xrow = X + (size_t)row * K;

    v8f acc = {0.f, 0.f, 0.f, 0.f, 0.f, 0.f, 0.f, 0.f};
#pragma unroll 8
    for (int k = 0; k < K; k += 4) {
        v2f a, b;
        a.x = xrow[k + koff];
        a.y = xrow[k + koff + 1];
        b.x = W[(k + koff)     * 64 + col];
        b.y = W[(k + koff + 1) * 64 + col];
        acc = __builtin_amdgcn_wmma_f32_16x16x4_f32(
            /*neg_a=*/false, a, /*neg_b=*/false, b,
            /*c_mod=*/(short)0, acc, /*reuse_a=*/false, /*reuse_b=*/false);
    }

    const int orow0 = blockIdx.x * 16 + half * 8;
#pragma unroll
    for (int r = 0; r < 8; ++r) {
        int orow = orow0 + r;
        if (orow < nrows) H[(size_t)orow * 64 + col] = acc[r];
    }
}

// ---------------- aggregation ----------------

// agg[i,c] = h[i,c] * dinv[i]^2   (self-loop term; also zero-initializes agg)
__global__ __launch_bounds__(256) void k_agg_init(const float* __restrict__ h,
                                                  const float* __restrict__ dinv,
                                                  float* __restrict__ agg, int n) {
    int t = blockIdx.x * blockDim.x + threadIdx.x;
    if (t < n * 64) {
        int i = t >> 6;
        float di = dinv[i];
        agg[t] = h[t] * di * di;
    }
}

// One wave per edge, 2 channels per lane: agg[dst,c] += h[src,c]*dinv[src]*dinv[dst]
__global__ __launch_bounds__(256) void k_agg_edges(const long long* __restrict__ ei,
                                                   const float* __restrict__ h,
                                                   const float* __restrict__ dinv,
                                                   float* __restrict__ agg, int E) {
    int t = blockIdx.x * blockDim.x + threadIdx.x;
    int e = t >> 5;
    if (e >= E) return;
    int c = (t & 31) * 2;
    long long s = ei[e];
    long long d = ei[(size_t)E + e];
    float w = dinv[s] * dinv[d];
    const float2 hv = *(const float2*)(h + (size_t)s * 64 + c);
    float* dstp = agg + (size_t)d * 64 + c;
    atomicAdd(dstp,     hv.x * w);   // global_atomic_add_f32, L2-resident target
    atomicAdd(dstp + 1, hv.y * w);
}

__global__ __launch_bounds__(256) void k_relu_bias(float* __restrict__ a,
                                                   const float* __restrict__ b, int n) {
    int t = blockIdx.x * blockDim.x + threadIdx.x;
    if (t < n * 64) a[t] = fmaxf(a[t] + b[t & 63], 0.0f);
}

__global__ __launch_bounds__(256) void k_bias(float* __restrict__ a,
                                              const float* __restrict__ b, int n) {
    int t = blockIdx.x * blockDim.x + threadIdx.x;
    if (t < n * 64) a[t] = a[t] + b[t & 63];
}

// ---------------- launch ----------------

extern "C" void kernel_launch(void* const* d_in, const int* in_sizes, int n_in,
                              void* d_out, int out_size, void* d_ws, size_t ws_size,
                              hipStream_t stream) {
    const float*     x  = (const float*)d_in[0];
    const long long* ei = (const long long*)d_in[1];   // [2, E] int64
    const float*     W1 = (const float*)d_in[2];
    const float*     b1 = (const float*)d_in[3];
    const float*     W2 = (const float*)d_in[4];
    const float*     b2 = (const float*)d_in[5];
    float* out = (float*)d_out;

    const int N = in_sizes[0] / IN_C;
    const int E = in_sizes[1] / 2;

    // workspace: dinv [N] (padded), hpre [N*64]
    float* dinv = (float*)d_ws;
    const size_t n_pad = ((size_t)N + 63) & ~(size_t)63;
    float* hpre = dinv + n_pad;

    const int T = 256;
    const int gN   = (N + T - 1) / T;
    const int gE   = (E + T - 1) / T;
    const int gNC  = (N * 64 + T - 1) / T;
    const int gEW  = (int)(((long long)E * 32 + T - 1) / T);
    const int gGem = (N + 15) / 16;

    // --- normalization: deg -> dinv ---
    k_fill1   <<<gN, T, 0, stream>>>(dinv, N);
    k_deg_edges<<<gE, T, 0, stream>>>(ei, dinv, E);
    k_rsqrt   <<<gN, T, 0, stream>>>(dinv, N);

    // --- layer 1: hpre = x @ W1 ; agg (in d_out) = A_norm @ hpre ; relu(+b1) ---
    k_gemm_wmma<IN_C><<<gGem, 128, 0, stream>>>(x, W1, hpre, N);
    k_agg_init <<<gNC, T, 0, stream>>>(hpre, dinv, out, N);
    k_agg_edges<<<gEW, T, 0, stream>>>(ei, hpre, dinv, out, E);
    k_relu_bias<<<gNC, T, 0, stream>>>(out, b1, N);

    // --- layer 2: hpre = out1 @ W2 ; d_out = A_norm @ hpre + b2 ---
    k_gemm_wmma<OUT1><<<gGem, 128, 0, stream>>>(out, W2, hpre, N);
    k_agg_init <<<gNC, T, 0, stream>>>(hpre, dinv, out, N);
    k_agg_edges<<<gEW, T, 0, stream>>>(ei, hpre, dinv, out, E);
    k_bias     <<<gNC, T, 0, stream>>>(out, b2, N);
}